// AttributeOperator_59425167507505
// MI455X (gfx1250) — compile-verified
//
#include <hip/hip_runtime.h>

#define D 512

typedef __attribute__((ext_vector_type(2))) float v2f;
typedef __attribute__((ext_vector_type(8))) float v8f;

// ---------------- grouping kernels ----------------

__global__ void zero_kernel(int* p, int n) {
  int i = blockIdx.x * blockDim.x + threadIdx.x;
  if (i < n) p[i] = 0;
}

__global__ void hist_kernel(const int* __restrict__ attrs, int nB, int* __restrict__ counts) {
  int i = blockIdx.x * blockDim.x + threadIdx.x;
  if (i < nB) atomicAdd(&counts[attrs[i]], 1);
}

__global__ void scan_kernel(const int* __restrict__ counts, int nAttrs, int* __restrict__ offsets) {
  if (blockIdx.x == 0 && threadIdx.x == 0) {
    int acc = 0;
    for (int a = 0; a < nAttrs; ++a) { offsets[a] = acc; acc += counts[a]; }
    offsets[nAttrs] = acc;
  }
}

__global__ void scatter_kernel(const int* __restrict__ attrs, int nB,
                               const int* __restrict__ offsets,
                               int* __restrict__ cursors, int* __restrict__ order) {
  int i = blockIdx.x * blockDim.x + threadIdx.x;
  if (i < nB) {
    int a = attrs[i];
    int pos = atomicAdd(&cursors[a], 1);
    order[offsets[a] + pos] = i;
  }
}

// ---------------- grouped WMMA GEMM ----------------
// grid.x = attribute id, grid.y = row-block (4 waves * 16 rows = 64 rows/block)
// Each wave computes a 16x16 fp32 tile: 16 output rows x up-to-16 samples
// sharing the same operator matrix, via V_WMMA_F32_16X16X4_F32.
// Invalid columns (group smaller than 16) read from a zeroed scratch vector,
// keeping the K loop completely branch-free (no exec-mask churn).

__global__ __launch_bounds__(128)
void attr_gemm_kernel(const float* __restrict__ attr_ops,
                      const float* __restrict__ obj_emb,
                      const int* __restrict__ objs,
                      const int* __restrict__ offsets,
                      const int* __restrict__ order,
                      const float* __restrict__ zvec,
                      float* __restrict__ out) {
  const int a = blockIdx.x;
  const int gStart = offsets[a];
  const int gCount = offsets[a + 1] - gStart;
  if (gCount == 0) return;  // block-uniform: safe w.r.t. WMMA EXEC requirement

  const int wave  = threadIdx.x >> 5;
  const int lane  = threadIdx.x & 31;
  const int half  = lane >> 4;      // 0 = lanes 0-15, 1 = lanes 16-31
  const int laneM = lane & 15;

  const int rowBase = (blockIdx.y * 4 + wave) * 16;
  const float* op   = attr_ops + (size_t)a * D * D;  // uniform base (SGPR)
  // A-matrix layout (16x4 fp32): lanes 0-15 VGPR{0,1} = K{0,1}; lanes 16-31 = K{2,3}
  const int aOff = (rowBase + laneM) * D + 2 * half; // 32-bit per-lane offset

  for (int n0 = 0; n0 < gCount; n0 += 16) {
    const int  nIdx  = n0 + laneM;
    const bool valid = nIdx < gCount;
    int bSample = 0;
    const float* repBase = zvec;  // invalid lanes stream zeros (cache-resident)
    if (valid) {
      bSample = order[gStart + nIdx];
      repBase = obj_emb + (size_t)objs[bSample] * D;
    }
    repBase += 2 * half;

    v8f c = {0.f, 0.f, 0.f, 0.f, 0.f, 0.f, 0.f, 0.f};
    for (int k = 0; k < D; k += 4) {
      // A tile: 8-byte aligned vector load of this lane's K pair
      v2f av = *(const v2f*)(op + aOff + k);
      // B tile (4x16): lane n holds B[K][n] = rep_n[K] for the matching K pair
      v2f bv = *(const v2f*)(repBase + k);
      // (neg_a, A, neg_b, B, c_mod, C, reuse_a, reuse_b)
      c = __builtin_amdgcn_wmma_f32_16x16x4_f32(
          false, av, false, bv, (short)0, c, false, false);
    }

    if (valid) {
      // C/D layout: VGPR r holds M = r (lanes 0-15) or r+8 (lanes 16-31), N = laneM
      float* o = out + (size_t)bSample * D + rowBase + 8 * half;
#pragma unroll
      for (int r = 0; r < 8; ++r) {
        float v = c[r];
        o[r] = v > 0.f ? v : 0.f;  // ReLU
      }
    }
  }
}

// ---------------- launch ----------------

extern "C" void kernel_launch(void* const* d_in, const int* in_sizes, int n_in,
                              void* d_out, int out_size, void* d_ws, size_t ws_size,
                              hipStream_t stream) {
  const int*   attrs    = (const int*)d_in[0];
  const int*   objs     = (const int*)d_in[1];
  const float* attr_ops = (const float*)d_in[2];
  const float* obj_emb  = (const float*)d_in[3];
  float*       out      = (float*)d_out;

  const int nB     = in_sizes[0];
  const int nAttrs = in_sizes[2] / (D * D);

  // workspace (ints): counts[nAttrs] | cursors[nAttrs] | offsets[nAttrs+1] |
  //                   order[nB] | (8B-aligned) zvec[D floats]
  int* counts  = (int*)d_ws;
  int* cursors = counts + nAttrs;
  int* offsets = cursors + nAttrs;
  int* order   = offsets + nAttrs + 1;
  int  zoffInts = 3 * nAttrs + 1 + nB;
  zoffInts = (zoffInts + 1) & ~1;                 // 8-byte align for v2f loads
  float* zvec  = (float*)((int*)d_ws + zoffInts); // D floats of zeros

  zero_kernel<<<(2 * nAttrs + 255) / 256, 256, 0, stream>>>(counts, 2 * nAttrs);
  zero_kernel<<<(D + 255) / 256, 256, 0, stream>>>((int*)zvec, D);
  hist_kernel<<<(nB + 255) / 256, 256, 0, stream>>>(attrs, nB, counts);
  scan_kernel<<<1, 32, 0, stream>>>(counts, nAttrs, offsets);
  scatter_kernel<<<(nB + 255) / 256, 256, 0, stream>>>(attrs, nB, offsets, cursors, order);

  dim3 grid(nAttrs, D / 64);  // 4 waves/block * 16 rows/wave = 64 rows
  attr_gemm_kernel<<<grid, 128, 0, stream>>>(attr_ops, obj_emb, objs, offsets, order, zvec, out);
}